// TripletLoss_41979010351722
// MI455X (gfx1250) — compile-verified
//
#include <hip/hip_runtime.h>

typedef float v2f __attribute__((ext_vector_type(2)));
typedef float v8f __attribute__((ext_vector_type(8)));

#define TL_MARGIN 0.5f
#define TL_EPS    1e-8f
#define TL_BIG    1e30f

// ---------------------------------------------------------------------------
// Kernel 1: squared norms per row (embeddings are 4MB -> L2 resident after this)
// ---------------------------------------------------------------------------
__global__ void tl_sqnorm_kernel(const float* __restrict__ E,
                                 float* __restrict__ sq, int N, int D) {
    int r = blockIdx.x * blockDim.x + threadIdx.x;
    if (r >= N) return;
    const float4* p = reinterpret_cast<const float4*>(E + (size_t)r * D);
    float s = 0.f;
    for (int k = 0; k < (D >> 2); ++k) {
        float4 v = p[k];
        s += v.x * v.x + v.y * v.y + v.z * v.z + v.w * v.w;
    }
    sq[r] = s;
}

// 16x16 fp32 Gram tile via V_WMMA_F32_16X16X4_F32, K = 128 (32 k-steps).
// A fragment (16x4): lane m=lane%16 holds row row0+m, v2f = K = {2h, 2h+1} of the
// 4-wide k-block (h = lane/16).  B fragment (4x16) has the identical per-lane
// pattern on the column block (Gram symmetry).  C/D: acc[v] <-> (i=row0+v+8h,
// j=col0+m).
//
// Four independent accumulators break the serial WMMA->WMMA RAW chain on the
// XDL pipe (only 2 waves/SIMD at 256 thr/WGP, so in-wave ILP matters).
__device__ __forceinline__ v8f tl_gram_tile(const v2f (&afr)[32],
                                            const float* __restrict__ brow) {
    v8f a0 = {0.f, 0.f, 0.f, 0.f, 0.f, 0.f, 0.f, 0.f};
    v8f a1 = a0, a2 = a0, a3 = a0;
#pragma unroll
    for (int kk = 0; kk < 32; kk += 4) {
        v2f b0 = *reinterpret_cast<const v2f*>(brow + 4 * (kk + 0));
        v2f b1 = *reinterpret_cast<const v2f*>(brow + 4 * (kk + 1));
        v2f b2 = *reinterpret_cast<const v2f*>(brow + 4 * (kk + 2));
        v2f b3 = *reinterpret_cast<const v2f*>(brow + 4 * (kk + 3));
        a0 = __builtin_amdgcn_wmma_f32_16x16x4_f32(false, afr[kk + 0], false, b0,
                                                   (short)0, a0, false, false);
        a1 = __builtin_amdgcn_wmma_f32_16x16x4_f32(false, afr[kk + 1], false, b1,
                                                   (short)0, a1, false, false);
        a2 = __builtin_amdgcn_wmma_f32_16x16x4_f32(false, afr[kk + 2], false, b2,
                                                   (short)0, a2, false, false);
        a3 = __builtin_amdgcn_wmma_f32_16x16x4_f32(false, afr[kk + 3], false, b3,
                                                   (short)0, a3, false, false);
    }
    return (a0 + a1) + (a2 + a3);
}

// ---------------------------------------------------------------------------
// Kernel 2: one workgroup per 16-row block; 8 waves stride over 16-col tiles.
// Pass 1: ap_sum / ap_count / hard-neg min.  Pass 2 (needs ap_mean): semi-hard
// min.  Per-row loss written to ws (no atomics -> deterministic).
// Assumes D == 128.
// ---------------------------------------------------------------------------
__launch_bounds__(256)
__global__ void tl_rows_kernel(const float* __restrict__ E,
                               const int* __restrict__ labels,
                               const float* __restrict__ sq,
                               float* __restrict__ loss, int N, int D) {
    const int tid  = threadIdx.x;
    const int wave = tid >> 5;        // 0..7
    const int lane = tid & 31;
    const int m    = lane & 15;
    const int h    = lane >> 4;
    const int row0 = blockIdx.x << 4;
    const int nTiles = N >> 4;
    constexpr int NW = 8;

    __shared__ float redA[NW][16];
    __shared__ float redC[NW][16];
    __shared__ float redH[NW][16];
    __shared__ float redS[NW][16];
    __shared__ float s_lo[16];
    __shared__ float s_hard[16];

    // Preload A fragments for the whole K range (64 VGPRs).
    v2f afr[32];
    const float* arow = E + (size_t)(row0 + m) * D + 2 * h;
#pragma unroll
    for (int kk = 0; kk < 32; ++kk)
        afr[kk] = *reinterpret_cast<const v2f*>(arow + 4 * kk);

    float sqI[8];
    int   lblI[8];
#pragma unroll
    for (int v = 0; v < 8; ++v) {
        int i   = row0 + v + 8 * h;
        sqI[v]  = sq[i];
        lblI[v] = labels[i];
    }

    // ---- Pass 1 -----------------------------------------------------------
    float apS[8], cntS[8], hardM[8];
#pragma unroll
    for (int v = 0; v < 8; ++v) { apS[v] = 0.f; cntS[v] = 0.f; hardM[v] = TL_BIG; }

    for (int ct = wave; ct < nTiles; ct += NW) {
        const int   col0 = ct << 4;
        const int   j    = col0 + m;
        const float sqJ  = sq[j];
        const int   lblJ = labels[j];
        v8f acc = tl_gram_tile(afr, E + (size_t)j * D + 2 * h);
#pragma unroll
        for (int v = 0; v < 8; ++v) {
            float d2 = fmaxf(sqI[v] + sqJ - 2.0f * acc[v], 0.f);
            float d  = sqrtf(d2);
            if (lblI[v] == lblJ) { apS[v] += d; cntS[v] += 1.f; }
            else                 { hardM[v] = fminf(hardM[v], d); }
        }
    }

    // Reduce across the 16 lanes of each half-wave.
#pragma unroll
    for (int v = 0; v < 8; ++v) {
#pragma unroll
        for (int off = 1; off < 16; off <<= 1) {
            apS[v]  += __shfl_xor(apS[v],  off, 32);
            cntS[v] += __shfl_xor(cntS[v], off, 32);
            hardM[v] = fminf(hardM[v], __shfl_xor(hardM[v], off, 32));
        }
    }
    if (m == 0) {
#pragma unroll
        for (int v = 0; v < 8; ++v) {
            int r = v + 8 * h;
            redA[wave][r] = apS[v];
            redC[wave][r] = cntS[v];
            redH[wave][r] = hardM[v];
        }
    }
    __syncthreads();
    if (tid < 16) {
        float a = 0.f, c = 0.f, hm = TL_BIG;
        for (int w = 0; w < NW; ++w) {
            a += redA[w][tid];
            c += redC[w][tid];
            hm = fminf(hm, redH[w][tid]);
        }
        s_lo[tid]   = a / (c + TL_EPS);
        s_hard[tid] = hm;
    }
    __syncthreads();

    // ---- Pass 2: semi-hard min in (lo, lo + MARGIN) over diff-label cols --
    float loV[8], shM[8];
#pragma unroll
    for (int v = 0; v < 8; ++v) { loV[v] = s_lo[v + 8 * h]; shM[v] = TL_BIG; }

    for (int ct = wave; ct < nTiles; ct += NW) {
        const int   col0 = ct << 4;
        const int   j    = col0 + m;
        const float sqJ  = sq[j];
        const int   lblJ = labels[j];
        v8f acc = tl_gram_tile(afr, E + (size_t)j * D + 2 * h);
#pragma unroll
        for (int v = 0; v < 8; ++v) {
            float d2 = fmaxf(sqI[v] + sqJ - 2.0f * acc[v], 0.f);
            float d  = sqrtf(d2);
            if ((lblI[v] != lblJ) && (d > loV[v]) && (d < loV[v] + TL_MARGIN))
                shM[v] = fminf(shM[v], d);
        }
    }

#pragma unroll
    for (int v = 0; v < 8; ++v) {
#pragma unroll
        for (int off = 1; off < 16; off <<= 1)
            shM[v] = fminf(shM[v], __shfl_xor(shM[v], off, 32));
    }
    if (m == 0) {
#pragma unroll
        for (int v = 0; v < 8; ++v) redS[wave][v + 8 * h] = shM[v];
    }
    __syncthreads();
    if (tid < 16) {
        float s = TL_BIG;
        for (int w = 0; w < NW; ++w) s = fminf(s, redS[w][tid]);
        // has_semi ? semi_hard_neg : hard_neg (sentinel TL_BIG clamps to 0 loss)
        float neg = (s < 0.5f * TL_BIG) ? s : s_hard[tid];
        loss[row0 + tid] = fmaxf(0.f, s_lo[tid] - neg + TL_MARGIN);
    }
}

// ---------------------------------------------------------------------------
// Kernel 3: deterministic single-block mean of per-row losses.
// ---------------------------------------------------------------------------
__global__ void tl_reduce_kernel(const float* __restrict__ loss,
                                 float* __restrict__ out, int N) {
    __shared__ float sm[256];
    float s = 0.f;
    for (int i = threadIdx.x; i < N; i += blockDim.x) s += loss[i];
    sm[threadIdx.x] = s;
    __syncthreads();
    for (int st = 128; st > 0; st >>= 1) {
        if ((int)threadIdx.x < st) sm[threadIdx.x] += sm[threadIdx.x + st];
        __syncthreads();
    }
    if (threadIdx.x == 0) out[0] = sm[0] / (float)N;
}

extern "C" void kernel_launch(void* const* d_in, const int* in_sizes, int n_in,
                              void* d_out, int out_size, void* d_ws, size_t ws_size,
                              hipStream_t stream) {
    const float* E      = (const float*)d_in[0];
    const int*   labels = (const int*)d_in[1];
    const int N = in_sizes[1];            // 8192
    const int D = in_sizes[0] / N;        // 128

    float* sq   = (float*)d_ws;           // N floats
    float* loss = sq + N;                 // N floats

    tl_sqnorm_kernel<<<(N + 255) / 256, 256, 0, stream>>>(E, sq, N, D);
    tl_rows_kernel<<<N / 16, 256, 0, stream>>>(E, labels, sq, loss, N, D);
    tl_reduce_kernel<<<1, 256, 0, stream>>>(loss, (float*)d_out, N);
}